// Vdw_46969762349275
// MI455X (gfx1250) — compile-verified
//
#include <hip/hip_runtime.h>

// ---------------------------------------------------------------------------
// LJ pair energy, MI455X (gfx1250, wave32).
// Pass 1: grid-stride over 10M pairs, coalesced int2 loads of the pair list
//         (the only real HBM stream, ~80 MB), L2-resident gathers for coords
//         and atom types, per-block partial sums to workspace.
// Pass 2: single-wave reduction of block partials using V_WMMA_F32_16X16X4_F32
//         as a 64-wide add tree (A = lane partials, B = ones).
// ---------------------------------------------------------------------------

typedef __attribute__((ext_vector_type(2))) float v2f;
typedef __attribute__((ext_vector_type(8))) float v8f;

#define THREADS 256
#define BLOCKS  2048
#define N_TYPES 32

__global__ __launch_bounds__(THREADS) void lj_pair_energy_kernel(
    const float* __restrict__ coords,
    const int2*  __restrict__ pairs,
    const float* __restrict__ box,
    const float* __restrict__ sigma,
    const float* __restrict__ epsil,
    const int*   __restrict__ cutoff,
    const int*   __restrict__ atypes,
    int npairs,
    float* __restrict__ partials)
{
    // Box is diagonal (BOX_L * I); read the diagonal once (scalar loads).
    const float Lx = box[0], Ly = box[4], Lz = box[8];
    const float iLx = 1.0f / Lx, iLy = 1.0f / Ly, iLz = 1.0f / Lz;
    const float rc  = (float)(*cutoff);
    const float rc2 = rc * rc;

    const int tid    = blockIdx.x * THREADS + threadIdx.x;
    const int stride = gridDim.x * THREADS;

    float acc = 0.0f;
    for (int i = tid; i < npairs; i += stride) {
        // gfx1250: global_prefetch_b8 for the next grid-stride chunk of the
        // sequential pair stream (speculative; OOB prefetches are dropped).
        __builtin_prefetch(pairs + i + stride, 0, 0);

        const int2 pr = pairs[i];
        const int ia = pr.x;
        const int ib = pr.y;

        const float ax = coords[ia * 3 + 0];
        const float ay = coords[ia * 3 + 1];
        const float az = coords[ia * 3 + 2];
        const float bx = coords[ib * 3 + 0];
        const float by = coords[ib * 3 + 1];
        const float bz = coords[ib * 3 + 2];

        const int ti = atypes[ia];
        const int tj = atypes[ib];
        const float sg = sigma[ti * N_TYPES + tj];
        const float ep = epsil[ti * N_TYPES + tj];

        // Minimum image; rintf = round-half-even, matching jnp.round.
        float dx = bx - ax; dx -= rintf(dx * iLx) * Lx;
        float dy = by - ay; dy -= rintf(dy * iLy) * Ly;
        float dz = bz - az; dz -= rintf(dz * iLz) * Lz;

        const float r2 = dx * dx + dy * dy + dz * dz;
        if (r2 <= rc2) {
            const float s2 = (sg * sg) / r2;      // (sigma/dr)^2, no sqrt
            const float t  = s2 * s2 * s2;        // (sigma/dr)^6
            acc += 4.0f * ep * t * (t - 1.0f);
        }
    }

    // Intra-wave reduction (wave32).
    #pragma unroll
    for (int off = 16; off > 0; off >>= 1)
        acc += __shfl_xor(acc, off, 32);

    __shared__ float wsum[THREADS / 32];
    const int wid = threadIdx.x >> 5;
    if ((threadIdx.x & 31) == 0) wsum[wid] = acc;
    __syncthreads();

    if (threadIdx.x == 0) {
        float t = 0.0f;
        #pragma unroll
        for (int k = 0; k < THREADS / 32; ++k) t += wsum[k];
        partials[blockIdx.x] = t;
    }
}

// Single-wave final reduction. Each lane serially folds partials into one
// float, then one V_WMMA_F32_16X16X4_F32 with an all-ones B matrix computes
// row-sums of the 16x4 A matrix holding the 32 lane partials:
//   A[m][0] = acc(lane m), A[m][2] = acc(lane m+16), other K = 0.
// D[m][n] = rowsum[m] replicated across columns; lanes 0-15 hold rows 0-7 in
// their 8 D VGPRs, lanes 16-31 hold rows 8-15, so
//   total = sum(c[0..7]) + shfl_xor(sum, 16).
__global__ __launch_bounds__(32) void lj_reduce_wmma_kernel(
    const float* __restrict__ partials, int n, float* __restrict__ out)
{
    const int lane = threadIdx.x;

    float acc = 0.0f;
    for (int i = lane; i < n; i += 32) acc += partials[i];

    v2f a; a[0] = acc;  a[1] = 0.0f;   // A matrix: 16x4 f32, 2 VGPRs/lane
    v2f b; b[0] = 1.0f; b[1] = 1.0f;   // B matrix: 4x16 all ones
    v8f c = {};
    c = __builtin_amdgcn_wmma_f32_16x16x4_f32(
        /*neg_a=*/false, a, /*neg_b=*/false, b,
        /*c_mod=*/(short)0, c, /*reuse_a=*/false, /*reuse_b=*/false);

    float s = c[0] + c[1] + c[2] + c[3] + c[4] + c[5] + c[6] + c[7];
    const float total = s + __shfl_xor(s, 16, 32);

    if (lane == 0) out[0] = total;
}

extern "C" void kernel_launch(void* const* d_in, const int* in_sizes, int n_in,
                              void* d_out, int out_size, void* d_ws, size_t ws_size,
                              hipStream_t stream) {
    // Inputs in setup_inputs() order:
    // 0: coords (f32, N_ATOMS*3)   1: pairs (int, N_PAIRS*2)
    // 2: box (f32, 9)              3: sigma (f32, 32*32)
    // 4: epsilon (f32, 32*32)      5: cutoff (int, 1)
    // 6: atom_types (int, N_ATOMS)
    const float* coords = (const float*)d_in[0];
    const int2*  pairs  = (const int2*)d_in[1];
    const float* box    = (const float*)d_in[2];
    const float* sigma  = (const float*)d_in[3];
    const float* epsil  = (const float*)d_in[4];
    const int*   cutoff = (const int*)d_in[5];
    const int*   atypes = (const int*)d_in[6];

    const int npairs = in_sizes[1] / 2;

    int blocks = BLOCKS;
    if (ws_size < (size_t)blocks * sizeof(float)) {
        blocks = (int)(ws_size / sizeof(float));
        if (blocks < 1) blocks = 1;
    }
    float* partials = (float*)d_ws;

    lj_pair_energy_kernel<<<blocks, THREADS, 0, stream>>>(
        coords, pairs, box, sigma, epsil, cutoff, atypes, npairs, partials);

    lj_reduce_wmma_kernel<<<1, 32, 0, stream>>>(partials, blocks, (float*)d_out);
}